// ScaledDotProductAttention_35227321761860
// MI455X (gfx1250) — compile-verified
//
#include <hip/hip_runtime.h>
#include <hip/hip_bf16.h>

typedef __attribute__((ext_vector_type(2))) float v2f;
typedef __attribute__((ext_vector_type(8))) float v8f;

#define NB    2
#define NH    16
#define NL    2048
#define ND    64
#define QROWS 32
#define SP    2052   // padded LDS row stride for S (floats): 4-bank shift/row
#define VST   72     // padded LDS row stride for V chunk: halves hit disjoint banks
#define VCHUNK 64    // K-rows of V staged per async chunk (16 KB payload)

__global__ __launch_bounds__(256)
void sdpa_fp32_wmma_kernel(const float* __restrict__ q,
                           const float* __restrict__ k,
                           const float* __restrict__ v,
                           const float* __restrict__ bias,
                           const unsigned char* __restrict__ mask,
                           float* __restrict__ outO,
                           float* __restrict__ outA)
{
    __shared__ float S[QROWS * SP];          // 262,656 B score strip
    __shared__ float Vb[2][VCHUNK * VST];    // 36,864 B double-buffered V stage
    __shared__ float partial[256];
    __shared__ float rowmax[QROWS];
    __shared__ float rinv[QROWS];

    const int tid  = threadIdx.x;
    const int wave = tid >> 5;
    const int lane = tid & 31;
    const int half = lane >> 4;              // 0: lanes 0-15, 1: lanes 16-31
    const int lr   = lane & 15;

    const int bh = blockIdx.x >> 6;          // 64 q-tiles per (b,h)
    const int qt = blockIdx.x & 63;
    const int q0 = qt * QROWS;
    const int bb = bh >> 4;                  // batch index (H = 16)

    const float* qp = q + (size_t)bh * NL * ND;
    const float* kp = k + (size_t)bh * NL * ND;
    const float* vp = v + (size_t)bh * NL * ND;
    const float* bp = bias + (size_t)bb * NL * NL;
    const unsigned char* mp = mask + (size_t)bb * NL * NL;

    // ---------------- Phase 1: S = Q K^T / 8 + bias, masked -> LDS ----------
    {
        const int wq = wave >> 2;            // 0..1 : q 16-row subtile
        const int wk = wave & 3;             // 0..3 : k column-tile group
        const int qrow = q0 + wq * 16 + lr;  // A-matrix: M = lane&15

        float ax[16], ay[16];                // A frags, 1/TEMPERATURE folded in
#pragma unroll
        for (int s = 0; s < 16; ++s) {
            const float2 a = *(const float2*)(qp + (size_t)qrow * ND + 4 * s + 2 * half);
            ax[s] = a.x * 0.125f;
            ay[s] = a.y * 0.125f;
        }

        for (int t = wk; t < NL / 16; t += 4) {
            const int n0   = t * 16;
            const int krow = n0 + lr;        // B-matrix: N = lane&15
            v8f c = {0.f, 0.f, 0.f, 0.f, 0.f, 0.f, 0.f, 0.f};
#pragma unroll
            for (int s = 0; s < 16; ++s) {
                const float2 bk = *(const float2*)(kp + (size_t)krow * ND + 4 * s + 2 * half);
                v2f A; A.x = ax[s]; A.y = ay[s];
                v2f Bv; Bv.x = bk.x; Bv.y = bk.y;
                c = __builtin_amdgcn_wmma_f32_16x16x4_f32(false, A, false, Bv,
                                                          (short)0, c, false, false);
            }
            if (t + 4 < NL / 16)             // prefetch next K column tile (near scope)
                __builtin_prefetch(kp + (size_t)(krow + 64) * ND, 0, 3);

            // C/D layout: element (M = j + 8*half, N = lane&15)
            const int col = n0 + lr;
#pragma unroll
            for (int j = 0; j < 8; ++j) {
                const int m    = j + 8 * half;
                const int grow = q0 + wq * 16 + m;
                float sv = c[j] + bp[(size_t)grow * NL + col];
                if (!mp[(size_t)grow * NL + col]) sv = -__builtin_huge_valf();
                S[(wq * 16 + m) * SP + col] = sv;
            }
        }
    }
    __syncthreads();

    // ---------------- Phase 2: exact softmax over the LDS strip -------------
    const int row = tid >> 3;                // 8 threads per row
    const int c0  = (tid & 7) * 256;         // 256 contiguous columns each

    float pm = -__builtin_huge_valf();
    for (int i = 0; i < 64; ++i) {
        const float4 x = *(const float4*)&S[row * SP + c0 + 4 * i];
        pm = fmaxf(pm, fmaxf(fmaxf(x.x, x.y), fmaxf(x.z, x.w)));
    }
    partial[tid] = pm;
    __syncthreads();
    if (tid < QROWS) {
        float m = partial[tid * 8];
#pragma unroll
        for (int j = 1; j < 8; ++j) m = fmaxf(m, partial[tid * 8 + j]);
        rowmax[tid] = m;
    }
    __syncthreads();

    {
        const float rm = rowmax[row];
        float sum = 0.f;
        for (int i = 0; i < 64; ++i) {
            float4 x = *(float4*)&S[row * SP + c0 + 4 * i];
            x.x = __expf(x.x - rm); x.y = __expf(x.y - rm);
            x.z = __expf(x.z - rm); x.w = __expf(x.w - rm);
            *(float4*)&S[row * SP + c0 + 4 * i] = x;
            sum += x.x + x.y + x.z + x.w;
        }
        partial[tid] = sum;
    }
    __syncthreads();
    if (tid < QROWS) {
        float s = 0.f;
#pragma unroll
        for (int j = 0; j < 8; ++j) s += partial[tid * 8 + j];
        rinv[tid] = 1.0f / s;
    }
    __syncthreads();

    // async stage of a 64x64 V chunk into LDS (ASYNCcnt-tracked DMA path)
    auto stage_v = [&](int c, int buf) {
#pragma unroll
        for (int i = 0; i < 4; ++i) {
            const int f  = tid + 256 * i;    // float4 index within chunk
            const int r  = f >> 4;           // chunk-local V row
            const int cq = (f & 15) * 4;     // column (float4 granule)
            const unsigned long long ga =
                (unsigned long long)(size_t)(vp + (size_t)(c * VCHUNK + r) * ND + cq);
            const unsigned loff = (unsigned)(size_t)
                (__attribute__((address_space(3))) float*)(&Vb[buf][r * VST + cq]);
            asm volatile("global_load_async_to_lds_b128 %0, %1, off"
                         :: "v"(loff), "v"(ga) : "memory");
        }
    };

    // kick off the first two V chunks; their latency hides behind the 256 KB
    // attention-probability writeback below
    stage_v(0, 0);
    stage_v(1, 1);

    // stream normalized probabilities to global (contiguous float4 stores)
    {
        const float rs = rinv[row];
        float* ap = outA + ((size_t)bh * NL + q0 + row) * NL + c0;
        for (int i = 0; i < 64; ++i) {
            float4 x = *(const float4*)&S[row * SP + c0 + 4 * i];
            x.x *= rs; x.y *= rs; x.z *= rs; x.w *= rs;
            *(float4*)(ap + 4 * i) = x;
        }
    }

    // ---------------- Phase 3: O = P @ V, V double-buffered via async DMA ---
    {
        const int qsub = wave >> 2;          // 0..1
        const int nsub = wave & 3;           // 0..3 -> 16-col slice of V
        const int vcol = nsub * 16 + lr;     // B-matrix: N = lane&15
        v8f acc = {0.f, 0.f, 0.f, 0.f, 0.f, 0.f, 0.f, 0.f};

        for (int c = 0; c < NL / VCHUNK; ++c) {
            // wait for chunk c (4 async loads/wave in-flight per chunk,
            // in-order completion), keep chunk c+1 streaming
            if (c < NL / VCHUNK - 1)
                asm volatile("s_wait_asynccnt 0x4" ::: "memory");
            else
                asm volatile("s_wait_asynccnt 0x0" ::: "memory");
            __syncthreads();                 // all waves' portions landed

            const int buf = c & 1;
#pragma unroll
            for (int s = 0; s < VCHUNK / 4; ++s) {
                const int kk = 4 * s + 2 * half;          // chunk-local K row
                const float2 a = *(const float2*)&S[(qsub * 16 + lr) * SP + c * VCHUNK + kk];
                v2f A; A.x = a.x; A.y = a.y;
                v2f Bv;
                Bv.x = Vb[buf][kk * VST + vcol];
                Bv.y = Vb[buf][(kk + 1) * VST + vcol];
                acc = __builtin_amdgcn_wmma_f32_16x16x4_f32(false, A, false, Bv,
                                                            (short)0, acc, false, false);
            }
            __syncthreads();                 // everyone done reading buf
            if (c + 2 < NL / VCHUNK)
                stage_v(c + 2, buf);         // refill the buffer just drained
        }

#pragma unroll
        for (int j = 0; j < 8; ++j) {
            const int m = j + 8 * half;
            outO[((size_t)bh * NL + q0 + qsub * 16 + m) * ND + vcol] =
                acc[j] * rinv[qsub * 16 + m];
        }
    }
}

extern "C" void kernel_launch(void* const* d_in, const int* in_sizes, int n_in,
                              void* d_out, int out_size, void* d_ws, size_t ws_size,
                              hipStream_t stream)
{
    const float*         q    = (const float*)d_in[0];
    const float*         k    = (const float*)d_in[1];
    const float*         v    = (const float*)d_in[2];
    const float*         bias = (const float*)d_in[3];
    const unsigned char* mask = (const unsigned char*)d_in[4];

    float* outO = (float*)d_out;                                   // (B,H,L,D)
    float* outA = (float*)d_out + (size_t)NB * NH * NL * ND;       // (B,H,L,L)

    const int grid = NB * NH * (NL / QROWS);   // 2048 workgroups
    sdpa_fp32_wmma_kernel<<<grid, 256, 0, stream>>>(q, k, v, bias, mask, outO, outA);
}